// TVLoss_7284264534672
// MI455X (gfx1250) — compile-verified
//
#include <hip/hip_runtime.h>
#include <hip/hip_bf16.h>
#include <math.h>

// ---------------------------------------------------------------------------
// TV loss on (1,3,4096,4096) f32.  Pure bandwidth problem (~201 MB @ 23.3 TB/s
// => ~9 us floor).  Stream with global_load_b128; reduce within each wave32
// using V_WMMA_F32_16X16X4_F32 (A = per-lane partials, B = ones), which
// collapses 32 lanes to two half-sums in a single matrix op.
// ---------------------------------------------------------------------------

typedef float v2f __attribute__((ext_vector_type(2)));
typedef float v8f __attribute__((ext_vector_type(8)));

#define C_PLANES   3
#define DIM        4096
#define BASE       4095               // rows/cols participating as "base"
#define CHUNKS_ROW 1024               // ceil(4095/4) 4-col chunks per row
#define TOTAL_CHUNKS (C_PLANES * BASE * CHUNKS_ROW)   // 12,579,840
#define NBLOCKS    2048
#define NTHREADS   256                // 8 waves of 32

// Full-wave (32 lane) sum via one WMMA.
// A-matrix 16x4 f32 layout (ISA 7.12.2): lane L<16 -> M=L, vgpr0=K0, vgpr1=K1;
// lane L>=16 -> M=L-16, vgpr0=K2, vgpr1=K3.  With A={acc,0} per lane and
// B = all ones:  D[m][n] = acc(m) + acc(m+16)  for every n.
// D layout: vgpr j = row j (lanes 0-15) / row j+8 (lanes 16-31).  Summing the
// 8 D vgprs gives rows 0-7 (lanes<16) or rows 8-15 (lanes>=16); xor-16 shuffle
// merges the halves so every lane ends with the full 32-lane sum.
__device__ __forceinline__ float wave_sum_wmma(float acc) {
    v2f a; a.x = acc;  a.y = 0.0f;
    v2f b; b.x = 1.0f; b.y = 1.0f;
    v8f c = {};
    v8f d = __builtin_amdgcn_wmma_f32_16x16x4_f32(
        /*neg_a=*/false, a, /*neg_b=*/false, b,
        /*c_mod=*/(short)0, c, /*reuse_a=*/false, /*reuse_b=*/false);
    float s = ((d[0] + d[1]) + (d[2] + d[3])) + ((d[4] + d[5]) + (d[6] + d[7]));
    s += __shfl_xor(s, 16, 32);
    return s;
}

__device__ __forceinline__ void block_reduce_store(float acc, float* dst) {
    __shared__ float red[NTHREADS / 32];
    const int lane = threadIdx.x & 31;
    const int wave = threadIdx.x >> 5;
    float w = wave_sum_wmma(acc);            // EXEC all-ones here: no divergence
    if (lane == 0) red[wave] = w;
    __syncthreads();
    if (threadIdx.x == 0) {
        float t = 0.0f;
        #pragma unroll
        for (int i = 0; i < NTHREADS / 32; ++i) t += red[i];
        *dst = t;
    }
}

__global__ void __launch_bounds__(NTHREADS)
tv_partial_kernel(const float* __restrict__ x, float* __restrict__ partials) {
    float acc = 0.0f;
    const int stride = gridDim.x * blockDim.x;
    for (int cid = blockIdx.x * blockDim.x + threadIdx.x;
         cid < TOTAL_CHUNKS; cid += stride) {
        const int c0  = (cid & (CHUNKS_ROW - 1)) << 2;        // 0..4092 step 4
        const int rr  = cid >> 10;                            // plane*4095+row
        const int row = rr % BASE;                            // 0..4094
        const int ch  = rr / BASE;                            // 0..2
        const size_t base = (size_t)ch * (DIM * DIM) + (size_t)row * DIM + c0;

        const float4 a = *(const float4*)(x + base);          // row r  (16B aligned)
        const float4 n = *(const float4*)(x + base + DIM);    // row r+1

        float dx0 = a.y - a.x, dy0 = n.x - a.x;
        float dx1 = a.z - a.y, dy1 = n.y - a.y;
        float dx2 = a.w - a.z, dy2 = n.z - a.z;
        acc = fmaf(dx0, dx0, acc); acc = fmaf(dy0, dy0, acc);
        acc = fmaf(dx1, dx1, acc); acc = fmaf(dy1, dy1, acc);
        acc = fmaf(dx2, dx2, acc); acc = fmaf(dy2, dy2, acc);

        if (c0 + 4 < DIM) {                                   // not the tail chunk
            float a4  = x[base + 4];
            float dx3 = a4  - a.w;
            float dy3 = n.w - a.w;
            acc = fmaf(dx3, dx3, acc);
            acc = fmaf(dy3, dy3, acc);
        }
    }
    block_reduce_store(acc, &partials[blockIdx.x]);
}

__global__ void __launch_bounds__(NTHREADS)
tv_final_kernel(const float* __restrict__ partials, float* __restrict__ out) {
    float acc = 0.0f;
    for (int i = threadIdx.x; i < NBLOCKS; i += NTHREADS) acc += partials[i];
    __shared__ float red[NTHREADS / 32];
    const int lane = threadIdx.x & 31;
    const int wave = threadIdx.x >> 5;
    float w = wave_sum_wmma(acc);
    if (lane == 0) red[wave] = w;
    __syncthreads();
    if (threadIdx.x == 0) {
        float t = 0.0f;
        #pragma unroll
        for (int i = 0; i < NTHREADS / 32; ++i) t += red[i];
        out[0] = sqrtf(t);
    }
}

extern "C" void kernel_launch(void* const* d_in, const int* in_sizes, int n_in,
                              void* d_out, int out_size, void* d_ws, size_t ws_size,
                              hipStream_t stream) {
    (void)in_sizes; (void)n_in; (void)out_size; (void)ws_size;
    const float* x  = (const float*)d_in[0];
    float* partials = (float*)d_ws;          // NBLOCKS floats (8 KB)
    float* out      = (float*)d_out;

    tv_partial_kernel<<<NBLOCKS, NTHREADS, 0, stream>>>(x, partials);
    tv_final_kernel<<<1, NTHREADS, 0, stream>>>(partials, out);
}